// MambaHAR_32186484916489
// MI455X (gfx1250) — compile-verified
//
#include <hip/hip_runtime.h>
#include <math.h>

// ---------------------------------------------------------------------------
// Mamba-HAR forward, MI455X (gfx1250, wave32).
// GEMMs use V_WMMA_F32_16X16X4_F32 (fp32 A/B/C) to match the fp32 reference.
// Big GEMMs stage W tiles through LDS (double-buffered, bank-conflict-free
// stride-20 layout) shared by the 4 waves of a block; 16 WMMAs issue
// back-to-back per K-chunk with prefetch of the next chunk overlapped.
// ---------------------------------------------------------------------------

#define BB   64
#define LL   512
#define FINN 9
#define DMM  256
#define NLAY 4
#define DINN 512
#define NST  16   // d_state
#define RRK  16   // dt_rank
#define KCV  4    // d_conv
#define NCLS 12
#define MROWS (BB * LL)      // 32768 token rows
#define XDW  (RRK + 2 * NST) // 48

#define KCHUNK 16            // K-depth staged per LDS chunk
#define LDST   20            // LDS row stride (words): conflict-free, 16B-aligned

typedef __attribute__((ext_vector_type(2))) float v2f;
typedef __attribute__((ext_vector_type(8))) float v8f;

static __device__ __forceinline__ v8f wmma_f32_16x16x4(v2f a, v2f b, v8f c) {
#if __has_builtin(__builtin_amdgcn_wmma_f32_16x16x4_f32)
    // (neg_a, A, neg_b, B, c_mod, C, reuse_a, reuse_b)
    return __builtin_amdgcn_wmma_f32_16x16x4_f32(false, a, false, b, (short)0, c,
                                                 false, false);
#else
    c[0] += a[0] * b[0];  // placeholder; histogram would show wmma=0
    return c;
#endif
}

// ---------------------------------------------------------------------------
// h = x @ inp_w^T + inp_b        (M x 9) x (256 x 9)^T -> (M x 256)
// ---------------------------------------------------------------------------
__global__ __launch_bounds__(256) void k_input_proj(
    const float* __restrict__ x, const float* __restrict__ w,
    const float* __restrict__ b, float* __restrict__ h) {
    const int m = blockIdx.x;
    const int o = threadIdx.x;
    const float* xr = x + (size_t)m * FINN;
    const float* wr = w + (size_t)o * FINN;
    float s = b[o];
#pragma unroll
    for (int f = 0; f < FINN; ++f) s += xr[f] * wr[f];
    h[(size_t)m * DMM + o] = s;
}

// ---------------------------------------------------------------------------
// LayerNorm over DM=256 per row; one 256-thread block per row.
// ---------------------------------------------------------------------------
__global__ __launch_bounds__(256) void k_layernorm(
    const float* __restrict__ h, const float* __restrict__ w,
    const float* __restrict__ b, float* __restrict__ out) {
    __shared__ float red[256];
    const int m = blockIdx.x, t = threadIdx.x;
    const float v = h[(size_t)m * DMM + t];
    red[t] = v;
    __syncthreads();
    for (int s = 128; s > 0; s >>= 1) {
        if (t < s) red[t] += red[t + s];
        __syncthreads();
    }
    const float mean = red[0] * (1.0f / DMM);
    __syncthreads();
    const float d = v - mean;
    red[t] = d * d;
    __syncthreads();
    for (int s = 128; s > 0; s >>= 1) {
        if (t < s) red[t] += red[t + s];
        __syncthreads();
    }
    const float var = red[0] * (1.0f / DMM);
    out[(size_t)m * DMM + t] = d * rsqrtf(var + 1e-5f) * w[t] + b[t];
}

// ---------------------------------------------------------------------------
// LDS-staged fp32 WMMA GEMM:  C[M x N] = A[M x K(lda)] * W[N x K]^T (+epilogue)
//   EPI : 0 = none, 1 = +bias then softplus, 2 = residual accumulate (C in/out)
// Block = 128 threads (4 row-waves) -> 64 rows x 64 cols per block.
// Grid  = (N/64, M/64); K % 16 == 0. All dims exact -> EXEC all ones (WMMA req).
// W slab (64 cols x 16 k) staged in LDS, double buffered, shared by all waves.
// fp32 WMMA layouts (wave32):
//   A 16x4 : lane m = lane&15, VGPR v holds K = v + 2*(lane>>4)
//   B 4x16 : lane n = lane&15, VGPR v holds K = v + 2*(lane>>4)
//   C 16x16: lane n = lane&15, VGPR v holds M = v + 8*(lane>>4)
// ---------------------------------------------------------------------------
template <int EPI>
__global__ __launch_bounds__(128) void k_gemm_wmma_lds(
    const float* __restrict__ A, int lda, const float* __restrict__ W, int K,
    const float* __restrict__ bias, float* __restrict__ C, int ldc) {
    __shared__ alignas(16) float sw[2][64 * LDST];

    const int tid  = threadIdx.x;
    const int lane = tid & 31;
    const int wid  = tid >> 5;
    const int hi   = lane >> 4;  // 0/1
    const int lo   = lane & 15;
    const int row0 = blockIdx.y * 64 + wid * 16;
    const int col0 = blockIdx.x * 64;

    // Cooperative staging mapping: 2 threads per W row, 8 floats each.
    const int sc = tid >> 1;            // 0..63 : column within slab
    const int sh = (tid & 1) * 8;       // 0 or 8 : k-offset within chunk
    const float* wg = W + (size_t)(col0 + sc) * K + sh;

    v8f acc[4];
#pragma unroll
    for (int j = 0; j < 4; ++j) {
        if constexpr (EPI == 2) {
#pragma unroll
            for (int v = 0; v < 8; ++v)
                acc[j][v] = C[(size_t)(row0 + v + 8 * hi) * ldc + col0 + j * 16 + lo];
        } else {
#pragma unroll
            for (int v = 0; v < 8; ++v) acc[j][v] = 0.0f;
        }
    }

    const float* arow = A + (size_t)(row0 + lo) * lda + 2 * hi;
    const int nch = K / KCHUNK;

    // ---- stage chunk 0 ----
    {
        const float4 w0 = *(const float4*)(wg);
        const float4 w1 = *(const float4*)(wg + 4);
        *(float4*)&sw[0][sc * LDST + sh]     = w0;
        *(float4*)&sw[0][sc * LDST + sh + 4] = w1;
    }
    v2f a_cur[4];
#pragma unroll
    for (int q = 0; q < 4; ++q) a_cur[q] = *(const v2f*)(arow + q * 4);
    __syncthreads();

    auto compute = [&](int buf) {
        const float* sb = &sw[buf][0];
        v2f bf[4][4];  // [k-step][col tile] : preload all fragments (ds bursts)
#pragma unroll
        for (int q = 0; q < 4; ++q)
#pragma unroll
            for (int j = 0; j < 4; ++j)
                bf[q][j] = *(const v2f*)&sb[(j * 16 + lo) * LDST + q * 4 + 2 * hi];
#pragma unroll
        for (int q = 0; q < 4; ++q)
#pragma unroll
            for (int j = 0; j < 4; ++j)  // independent acc chains across j
                acc[j] = wmma_f32_16x16x4(a_cur[q], bf[q][j], acc[j]);
    };

    for (int c = 0; c < nch - 1; ++c) {
        // prefetch chunk c+1 (global -> regs), overlapped with compute(c)
        const int k0 = (c + 1) * KCHUNK;
        const float4 wn0 = *(const float4*)(wg + k0);
        const float4 wn1 = *(const float4*)(wg + k0 + 4);
        v2f a_nxt[4];
#pragma unroll
        for (int q = 0; q < 4; ++q) a_nxt[q] = *(const v2f*)(arow + k0 + q * 4);

        compute(c & 1);

        __syncthreads();  // everyone done reading buf[(c+1)&1] (used at c-1)
        *(float4*)&sw[(c + 1) & 1][sc * LDST + sh]     = wn0;
        *(float4*)&sw[(c + 1) & 1][sc * LDST + sh + 4] = wn1;
#pragma unroll
        for (int q = 0; q < 4; ++q) a_cur[q] = a_nxt[q];
        __syncthreads();  // stores visible before compute(c+1)
    }
    compute((nch - 1) & 1);

    // ---- epilogue ----
#pragma unroll
    for (int j = 0; j < 4; ++j) {
        const int col = col0 + j * 16 + lo;
        float bv = 0.0f;
        if constexpr (EPI == 1) bv = bias[col];
#pragma unroll
        for (int v = 0; v < 8; ++v) {
            float r = acc[j][v];
            if constexpr (EPI == 1) {
                r += bv;
                r = (r > 20.0f) ? r : log1pf(__expf(r));  // softplus
            }
            C[(size_t)(row0 + v + 8 * hi) * ldc + col] = r;
        }
    }
}

// ---------------------------------------------------------------------------
// Direct-global fp32 WMMA GEMM with register pipelining (for N=48 x-proj).
// Wave tile = 16 rows x 16*TN cols; block = 4 row-waves.
// ---------------------------------------------------------------------------
template <int TN, int EPI>
__global__ __launch_bounds__(128) void k_gemm_wmma_g(
    const float* __restrict__ A, int lda, const float* __restrict__ W, int K,
    const float* __restrict__ bias, float* __restrict__ C, int ldc) {
    const int lane = threadIdx.x & 31;
    const int wid  = threadIdx.x >> 5;
    const int hi   = lane >> 4;
    const int lo   = lane & 15;
    const int row0 = blockIdx.y * 64 + wid * 16;
    const int col0 = blockIdx.x * (16 * TN);

    v8f acc[TN];
#pragma unroll
    for (int j = 0; j < TN; ++j) {
        if constexpr (EPI == 2) {
#pragma unroll
            for (int v = 0; v < 8; ++v)
                acc[j][v] = C[(size_t)(row0 + v + 8 * hi) * ldc + col0 + j * 16 + lo];
        } else {
#pragma unroll
            for (int v = 0; v < 8; ++v) acc[j][v] = 0.0f;
        }
    }

    const float* arow = A + (size_t)(row0 + lo) * lda + 2 * hi;
    const float* wr[TN];
#pragma unroll
    for (int j = 0; j < TN; ++j)
        wr[j] = W + (size_t)(col0 + j * 16 + lo) * K + 2 * hi;

    v2f a_cur = *(const v2f*)(arow);
    v2f b_cur[TN];
#pragma unroll
    for (int j = 0; j < TN; ++j) b_cur[j] = *(const v2f*)(wr[j]);

    for (int k = 4; k < K; k += 4) {
        const v2f a_n = *(const v2f*)(arow + k);  // prefetch before WMMAs
        v2f b_n[TN];
#pragma unroll
        for (int j = 0; j < TN; ++j) b_n[j] = *(const v2f*)(wr[j] + k);
#pragma unroll
        for (int j = 0; j < TN; ++j)
            acc[j] = wmma_f32_16x16x4(a_cur, b_cur[j], acc[j]);
        a_cur = a_n;
#pragma unroll
        for (int j = 0; j < TN; ++j) b_cur[j] = b_n[j];
    }
#pragma unroll
    for (int j = 0; j < TN; ++j)
        acc[j] = wmma_f32_16x16x4(a_cur, b_cur[j], acc[j]);

#pragma unroll
    for (int j = 0; j < TN; ++j) {
        const int col = col0 + j * 16 + lo;
        float bv = 0.0f;
        if constexpr (EPI == 1) bv = bias[col];
#pragma unroll
        for (int v = 0; v < 8; ++v) {
            float r = acc[j][v];
            if constexpr (EPI == 1) {
                r += bv;
                r = (r > 20.0f) ? r : log1pf(__expf(r));
            }
            C[(size_t)(row0 + v + 8 * hi) * ldc + col] = r;
        }
    }
}

// ---------------------------------------------------------------------------
// Depthwise causal conv (K=4) over time + SiLU.
// ---------------------------------------------------------------------------
__global__ __launch_bounds__(256) void k_conv_silu(
    const float* __restrict__ xz, const float* __restrict__ cw,
    const float* __restrict__ cb, float* __restrict__ xc) {
    const int idx = blockIdx.x * 256 + threadIdx.x;  // (b, t, d) flat
    const int d = idx & (DINN - 1);
    const int t = (idx >> 9) & (LL - 1);
    const int b = idx >> 18;
    const float* w = cw + d * KCV;
    float s = cb[d];
#pragma unroll
    for (int j = 0; j < KCV; ++j) {
        const int tt = t - (KCV - 1) + j;
        if (tt >= 0) s += w[j] * xz[(size_t)(b * LL + tt) * (2 * DINN) + d];
    }
    xc[(size_t)idx] = s / (1.0f + __expf(-s));  // silu
}

// ---------------------------------------------------------------------------
// Selective scan fused with +u*D and *silu(z); one thread per (b,d) lane,
// 16 fp32 states in registers; writes y in place over u (xc buffer).
// ---------------------------------------------------------------------------
__global__ __launch_bounds__(256) void k_selective_scan(
    const float* __restrict__ dt, const float* __restrict__ xdbl,
    const float* __restrict__ Alog, const float* __restrict__ Dp,
    const float* __restrict__ xz, float* __restrict__ u_y) {
    const int idx = blockIdx.x * 256 + threadIdx.x;  // b*DIN + d
    const int d = idx & (DINN - 1);
    const int b = idx >> 9;
    float a[NST], h[NST];
#pragma unroll
    for (int n = 0; n < NST; ++n) {
        a[n] = -__expf(Alog[d * NST + n]);  // A = -exp(A_log)
        h[n] = 0.0f;
    }
    const float Dd = Dp[d];
    for (int t = 0; t < LL; ++t) {
        const size_t r = (size_t)(b * LL + t);
        const float uv  = u_y[r * DINN + d];
        const float dtv = dt[r * DINN + d];
        const float du  = dtv * uv;
        const float* bc = xdbl + r * XDW;  // row: [dt(16) | B(16) | C(16)]
        float y = 0.0f;
#pragma unroll
        for (int n = 0; n < NST; ++n) {
            const float dA = __expf(dtv * a[n]);
            h[n] = h[n] * dA + du * bc[RRK + n];
            y += h[n] * bc[RRK + NST + n];
        }
        const float zv = xz[r * (2 * DINN) + DINN + d];
        const float sz = zv / (1.0f + __expf(-zv));  // silu(z)
        u_y[r * DINN + d] = (y + uv * Dd) * sz;
    }
}

// ---------------------------------------------------------------------------
// Mean-pool over L then fc.
// ---------------------------------------------------------------------------
__global__ __launch_bounds__(256) void k_pool_fc(
    const float* __restrict__ h, const float* __restrict__ fw,
    const float* __restrict__ fb, float* __restrict__ out) {
    __shared__ float pooled[DMM];
    const int b = blockIdx.x, t = threadIdx.x;
    float s = 0.0f;
    for (int l = 0; l < LL; ++l) s += h[(size_t)(b * LL + l) * DMM + t];
    pooled[t] = s * (1.0f / LL);
    __syncthreads();
    if (t < NCLS) {
        float o = fb[t];
        for (int k = 0; k < DMM; ++k) o += pooled[k] * fw[t * DMM + k];
        out[b * NCLS + t] = o;
    }
}

// ---------------------------------------------------------------------------
extern "C" void kernel_launch(void* const* d_in, const int* in_sizes, int n_in,
                              void* d_out, int out_size, void* d_ws,
                              size_t ws_size, hipStream_t stream) {
    const float* x         = (const float*)d_in[0];
    const float* inp_w     = (const float*)d_in[1];
    const float* inp_b     = (const float*)d_in[2];
    const float* ln_w      = (const float*)d_in[3];
    const float* ln_b      = (const float*)d_in[4];
    const float* inproj_w  = (const float*)d_in[5];
    const float* conv_w    = (const float*)d_in[6];
    const float* conv_b    = (const float*)d_in[7];
    const float* xproj_w   = (const float*)d_in[8];
    const float* dtproj_w  = (const float*)d_in[9];
    const float* dtproj_b  = (const float*)d_in[10];
    const float* A_log     = (const float*)d_in[11];
    const float* Dp        = (const float*)d_in[12];
    const float* outproj_w = (const float*)d_in[13];
    const float* fc_w      = (const float*)d_in[14];
    const float* fc_b      = (const float*)d_in[15];

    // Workspace layout (floats); total ~85.5M floats ~= 342 MB.
    float* ws   = (float*)d_ws;
    float* h    = ws;                               // M x 256
    float* hn   = h    + (size_t)MROWS * DMM;       // M x 256
    float* xz   = hn   + (size_t)MROWS * DMM;       // M x 1024  (xi | z)
    float* xc   = xz   + (size_t)MROWS * 2 * DINN;  // M x 512   (u, then y)
    float* xdbl = xc   + (size_t)MROWS * DINN;      // M x 48    (dt|B|C)
    float* dtb  = xdbl + (size_t)MROWS * XDW;       // M x 512

    k_input_proj<<<MROWS, 256, 0, stream>>>(x, inp_w, inp_b, h);

    for (int l = 0; l < NLAY; ++l) {
        k_layernorm<<<MROWS, 256, 0, stream>>>(h, ln_w + l * DMM, ln_b + l * DMM, hn);

        // xz = hn @ inproj_w[l]^T : (M,256)x(1024,256)^T -> (M,1024)
        k_gemm_wmma_lds<0><<<dim3((2 * DINN) / 64, MROWS / 64), 128, 0, stream>>>(
            hn, DMM, inproj_w + (size_t)l * 2 * DINN * DMM, DMM, nullptr, xz,
            2 * DINN);

        k_conv_silu<<<(MROWS * DINN) / 256, 256, 0, stream>>>(
            xz, conv_w + l * DINN * KCV, conv_b + l * DINN, xc);

        // xdbl = xc @ xproj_w[l]^T : (M,512)x(48,512)^T -> (M,48)
        k_gemm_wmma_g<3, 0><<<dim3(1, MROWS / 64), 128, 0, stream>>>(
            xc, DINN, xproj_w + (size_t)l * XDW * DINN, DINN, nullptr, xdbl, XDW);

        // dt = softplus(xdbl[:, :16] @ dtproj_w[l]^T + dtproj_b[l]) -> (M,512)
        k_gemm_wmma_lds<1><<<dim3(DINN / 64, MROWS / 64), 128, 0, stream>>>(
            xdbl, XDW, dtproj_w + (size_t)l * DINN * RRK, RRK,
            dtproj_b + l * DINN, dtb, DINN);

        k_selective_scan<<<(BB * DINN) / 256, 256, 0, stream>>>(
            dtb, xdbl, A_log + (size_t)l * DINN * NST, Dp + l * DINN, xz, xc);

        // h += y @ outproj_w[l]^T : (M,512)x(256,512)^T -> (M,256), residual
        // fused by preloading the WMMA C operand with the existing h tile.
        k_gemm_wmma_lds<2><<<dim3(DMM / 64, MROWS / 64), 128, 0, stream>>>(
            xc, DINN, outproj_w + (size_t)l * DMM * DINN, DINN, nullptr, h, DMM);
    }

    k_pool_fc<<<BB, 256, 0, stream>>>(h, fc_w, fc_b, (float*)d_out);
}